// DiscretizationLayer_59828894433825
// MI455X (gfx1250) — compile-verified
//
#include <hip/hip_runtime.h>

// ZOH discretization: A_d = sum_{i=0..10} (A*dt)^i / i!
//                     B_d = dt*B + (sum_{i=1..9} dt^{i+1}/10! * (A*dt)^i) @ B
// One workgroup (512 threads = 16 wave32) per batch matrix; all matmuls via
// v_wmma_f32_16x16x4_f32. B-operand fragments of the power chain (always sAt)
// are cached in registers across all 9 iterations to halve LDS traffic.

#define NDIM 64
#define LSTR 68   // padded LDS row stride (floats): 272B rows -> 16B aligned, conflict-free strided reads

typedef float v2f __attribute__((ext_vector_type(2)));
typedef float v4f __attribute__((ext_vector_type(4)));
typedef float v8f __attribute__((ext_vector_type(8)));

// B-frag layout (4x16 f32): v0 -> rows {0,2} split across lane halves, v1 -> rows {1,3}.
__device__ __forceinline__ void load_bfrags(const float* __restrict__ Q,
                                            int bcol, int half, int l16,
                                            v2f bf[16])
{
#pragma unroll
  for (int kk = 0; kk < 16; ++kk) {
    const int k4 = kk * 4;
    bf[kk].x = Q[(k4 + 2 * half + 0) * LSTR + bcol + l16];
    bf[kk].y = Q[(k4 + 2 * half + 1) * LSTR + bcol + l16];
  }
}

// C(16x16) = P[arow..+15][0..63] * Q where Q's fragments are pre-loaded in bf[].
// A-frag layout (16x4 f32): lanes 0-15 hold K=0,1 in v[0],v[1]; lanes 16-31 hold K=2,3.
__device__ __forceinline__ v8f tile_mm64(const float* __restrict__ P,
                                         int arow, int half, int l16,
                                         const v2f bf[16])
{
  v8f c = {};
#pragma unroll
  for (int kk = 0; kk < 16; ++kk) {
    v2f a = *(const v2f*)&P[(arow + l16) * LSTR + kk * 4 + 2 * half]; // ds_load_b64
    // (neg_a, A, neg_b, B, c_mod, C, reuse_a, reuse_b)
    c = __builtin_amdgcn_wmma_f32_16x16x4_f32(false, a, false, bf[kk], (short)0,
                                              c, false, false);
  }
  return c;
}

__global__ __launch_bounds__(512, 1)
void zoh_discretize_wmma(const float* __restrict__ Ag,
                         const float* __restrict__ Bg,
                         const float* __restrict__ pg,
                         float* __restrict__ Adg,
                         float* __restrict__ Bdg)
{
  __shared__ float sAt[NDIM * LSTR];
  __shared__ float sB [NDIM * LSTR];
  __shared__ float sP0[NDIM * LSTR];
  __shared__ float sP1[NDIM * LSTR];

  const int tid  = threadIdx.x;
  const int lane = tid & 31;
  const int wv   = tid >> 5;        // 0..15 waves
  const int ti   = (wv >> 2) * 16;  // output tile row base
  const int tj   = (wv & 3) * 16;   // output tile col base
  const int half = lane >> 4;       // 0: lanes 0-15, 1: lanes 16-31
  const int l16  = lane & 15;

  const size_t mat = (size_t)blockIdx.x * (NDIM * NDIM);
  const float pv = pg[0];
  const float dt = 1.0f / (1.0f + __expf(-pv)) * (0.1f - 1e-4f) + 1e-4f;

  // Stage A*dt and B into padded LDS (2x float4 per thread, coalesced b128).
  {
    const float* gA = Ag + mat;
    const float* gB = Bg + mat;
#pragma unroll
    for (int k = 0; k < 2; ++k) {
      const int e4  = tid + k * 512;   // float4 index, 0..1023
      const int row = e4 >> 4;         // 16 float4 per 64-float row
      const int c4  = (e4 & 15) * 4;
      v4f a = *(const v4f*)(gA + e4 * 4);
      v4f b = *(const v4f*)(gB + e4 * 4);
      a *= dt;
      *(v4f*)&sAt[row * LSTR + c4] = a;
      *(v4f*)&sB [row * LSTR + c4] = b;
    }
  }
  __syncthreads();

  const float inv10f = 1.0f / 3628800.0f;   // 1/10!

  // Cache this wave's B-operand fragments of At once: reused by all 9 power matmuls.
  v2f bfAt[16];
  load_bfrags(sAt, tj, half, l16, bfAt);

  // Register accumulators: A_d tile and S = sum c_i*At^i tile (C/D WMMA layout).
  v8f accA, accS;
#pragma unroll
  for (int r = 0; r < 8; ++r) {
    const int row = ti + r + half * 8;
    const int col = tj + l16;
    const float at = sAt[row * LSTR + col];
    accA[r] = (row == col ? 1.0f : 0.0f) + at;   // i=0 (I) and i=1 (At/1!)
    accS[r] = (dt * dt * inv10f) * at;           // c_1 * At
  }

  // Power chain: At^i = At^{i-1} @ At, ping-pong in LDS, 1 barrier per term.
  const float* pread  = sAt;
  float*       pwrite = sP0;
  float fact  = 1.0f;      // i!      (folds to literals under full unroll)
  float dtpow = dt * dt;   // dt^{i+1}
#pragma unroll
  for (int i = 2; i <= 10; ++i) {
    v8f T = tile_mm64(pread, ti, half, l16, bfAt);   // T = At^i tile
    fact  *= (float)i;
    dtpow *= dt;
    const float fa = 1.0f / fact;
    const float fs = dtpow * inv10f;
#pragma unroll
    for (int r = 0; r < 8; ++r) {
      accA[r] += T[r] * fa;
      if (i <= 9) accS[r] += T[r] * fs;
    }
    if (i < 10) {
#pragma unroll
      for (int r = 0; r < 8; ++r)
        pwrite[(ti + r + half * 8) * LSTR + tj + l16] = T[r];
      __syncthreads();
      pread  = pwrite;
      pwrite = (pwrite == sP0) ? sP1 : sP0;
    }
  }

  // B_d = dt*B + S @ B : dump S into the free ping-pong buffer, one matmul.
#pragma unroll
  for (int r = 0; r < 8; ++r)
    pwrite[(ti + r + half * 8) * LSTR + tj + l16] = accS[r];
  __syncthreads();

  v2f bfB[16];
  load_bfrags(sB, tj, half, l16, bfB);
  v8f TB = tile_mm64(pwrite, ti, half, l16, bfB);

  float* oA = Adg + mat;
  float* oB = Bdg + mat;
#pragma unroll
  for (int r = 0; r < 8; ++r) {
    const int row = ti + r + half * 8;
    const int col = tj + l16;
    oA[row * NDIM + col] = accA[r];
    oB[row * NDIM + col] = dt * sB[row * LSTR + col] + TB[r];
  }
}

extern "C" void kernel_launch(void* const* d_in, const int* in_sizes, int n_in,
                              void* d_out, int out_size, void* d_ws, size_t ws_size,
                              hipStream_t stream) {
  const float* A = (const float*)d_in[0];
  const float* B = (const float*)d_in[1];
  const float* p = (const float*)d_in[2];
  const int batch = in_sizes[0] / (NDIM * NDIM);

  float* Ad = (float*)d_out;
  float* Bd = Ad + (size_t)batch * NDIM * NDIM;

  zoh_discretize_wmma<<<batch, 512, 0, stream>>>(A, B, p, Ad, Bd);
}